// InstanceRouteOptimizationArea_42700564857382
// MI455X (gfx1250) — compile-verified
//
#include <hip/hip_runtime.h>

// ---------------------------------------------------------------------------
// InstanceRouteOptimizationArea for MI455X (gfx1250, wave32)
//
// map_x/map_y = (ox*c)^T @ oy : two GEMMs M=128,N=128,K=100000 done with
// V_WMMA_F32_16X16X4_F32, fragments generated on the fly from net bboxes.
// One wave computes BOTH maps for one 16x16 tile (shared ox/oy fragment
// build, 2 WMMAs per K-step). K padded to a multiple of 4 -> branch-free
// inner loop, EXEC all-ones (WMMA requirement). Partials per K-chunk are
// plain stores, reduced in fixed order (deterministic, no float atomics).
// Kernel 4 stages the util map into LDS via global_load_async_to_lds_b128.
// ---------------------------------------------------------------------------

namespace {
constexpr int   kNumBins    = 128;
constexpr int   kNumNets    = 100000;
constexpr int   kNumNodes   = 200000;
constexpr int   kNumMovable = 180000;
constexpr int   kNumPins    = 400000;
constexpr float kBinS       = 1000.0f / 128.0f;       // 7.8125
constexpr float kBinArea    = kBinS * kBinS;
constexpr float kInvHCap    = 1.0f / (kBinArea * 1.5f);
constexpr float kInvVCap    = 1.0f / (kBinArea * 1.4f);
constexpr int   kChunks     = 64;                      // K-split for parallelism
constexpr int   kChunkLen   = 1564;                    // multiple of 4
constexpr int   kPadNets    = kChunks * kChunkLen;     // 100096 (zero-padded)
constexpr int   kMapElems   = kNumBins * kNumBins;     // 16384
}

typedef __attribute__((ext_vector_type(2))) float v2f;
typedef __attribute__((ext_vector_type(8))) float v8f;

#if defined(__gfx1250__) && __has_builtin(__builtin_amdgcn_wmma_f32_16x16x4_f32)
#define USE_WMMA_F32X4 1
#else
#define USE_WMMA_F32X4 0
#endif

// ---------------------------------------------------------------------------
// Kernel 1: per-net bbox + weights, zero-padded to kPadNets.
// netf[8*n] = {xmin,xmax,ymin,ymax,cx,cy,0,0}; pad entries are all-zero
// (=> zero fragments, contribute nothing to the GEMM).
// ---------------------------------------------------------------------------
__global__ __launch_bounds__(256) void net_bbox_kernel(
    const float* __restrict__ pin_pos,
    const int*   __restrict__ netpin_start,
    const int*   __restrict__ flat_netpin,
    const float* __restrict__ net_weights,
    float*       __restrict__ netf) {
  const int n = blockIdx.x * blockDim.x + threadIdx.x;
  if (n >= kPadNets) return;
  float4 bb = {0.f, 0.f, 0.f, 0.f};
  float2 cc = {0.f, 0.f};
  if (n < kNumNets) {
    const int s = netpin_start[n];
    const int e = netpin_start[n + 1];
    float xmin = 3.4e38f, xmax = -3.4e38f, ymin = 3.4e38f, ymax = -3.4e38f;
    for (int i = s; i < e; ++i) {
      const int p = flat_netpin[i];
      const float px = pin_pos[p];
      const float py = pin_pos[p + kNumPins];
      xmin = fminf(xmin, px); xmax = fmaxf(xmax, px);
      ymin = fminf(ymin, py); ymax = fmaxf(ymax, py);
    }
    const float w  = net_weights[n];
    const float dx = xmax - xmin;
    const float dy = ymax - ymin;
    bb = make_float4(xmin, xmax, ymin, ymax);
    cc = make_float2((dy > 0.f) ? (w / dy) : 0.f,    // cx (map_x scale)
                     (dx > 0.f) ? (w / dx) : 0.f);   // cy (map_y scale)
  }
  float* o = netf + 8 * n;
  *(float4*)(o)     = bb;
  *(float2*)(o + 4) = cc;
  o[6] = 0.f; o[7] = 0.f;
}

// ---------------------------------------------------------------------------
// Kernel 2: WMMA rank-K accumulation, both maps per wave.
//   grid = kChunks * 8 blocks of 256 threads (8 waves); wave -> (tile, chunk)
//   tile in [0,64): tx = x-tile, ty = y-tile of the 16x16 output tile.
// A-frag (16x4 f32): lanes 0-15 M=0..15 {VGPR0:K=0, VGPR1:K=1},
//                    lanes 16-31 M=0..15 {VGPR0:K=2, VGPR1:K=3}; B mirrored.
// C/D: VGPR r = rows {r, r+8}, N = lane%16.
// ---------------------------------------------------------------------------
__global__ __launch_bounds__(256) void routemap_kernel(
    const float* __restrict__ netf,
    float*       __restrict__ partial) {
  const int lane  = threadIdx.x & 31;
  const int wave  = threadIdx.x >> 5;
  const int chunk = blockIdx.x >> 3;                  // [0, kChunks)
  const int tile  = ((blockIdx.x & 7) << 3) | wave;   // [0, 64)
  const int tx    = tile & 7;                         // x-bin tile (rows)
  const int ty    = tile >> 3;                        // y-bin tile (cols)
  const int half  = lane >> 4;
  const int lm    = lane & 15;

  const float bxl = (float)(tx * 16 + lm) * kBinS;
  const float bxr = bxl + kBinS;
  const float byl = (float)(ty * 16 + lm) * kBinS;
  const float byr = byl + kBinS;

  const int kbeg = chunk * kChunkLen;
  const int kend = kbeg + kChunkLen;    // always in-bounds (padded netf)

  v8f accx = {0.f, 0.f, 0.f, 0.f, 0.f, 0.f, 0.f, 0.f};
  v8f accy = {0.f, 0.f, 0.f, 0.f, 0.f, 0.f, 0.f, 0.f};

#pragma unroll 2
  for (int k0 = kbeg; k0 < kend; k0 += 4) {
#if USE_WMMA_F32X4
    v2f ax, ay, b;
#pragma unroll
    for (int j = 0; j < 2; ++j) {
      const int n = k0 + half * 2 + j;   // K index this VGPR holds
      const float4 bb = *(const float4*)(netf + 8 * n);  // xmin,xmax,ymin,ymax
      const float2 cc = *(const float2*)(netf + 8 * n + 4);
      const float ox = fmaxf(fminf(bxr, bb.y) - fmaxf(bxl, bb.x), 0.f);
      const float oy = fmaxf(fminf(byr, bb.w) - fmaxf(byl, bb.z), 0.f);
      ax[j] = ox * cc.x;
      ay[j] = ox * cc.y;
      b[j]  = oy;
    }
    accx = __builtin_amdgcn_wmma_f32_16x16x4_f32(
        false, ax, false, b, (short)0, accx, false, false);
    accy = __builtin_amdgcn_wmma_f32_16x16x4_f32(
        false, ay, false, b, (short)0, accy, false, false);
#else
    // VALU fallback: lane computes its 8 C elements of both maps directly.
#pragma unroll
    for (int j = 0; j < 4; ++j) {
      const int n = k0 + j;
      const float4 bb = *(const float4*)(netf + 8 * n);
      const float2 cc = *(const float2*)(netf + 8 * n + 4);
      const float oy  = fmaxf(fminf(byr, bb.w) - fmaxf(byl, bb.z), 0.f);
#pragma unroll
      for (int r = 0; r < 8; ++r) {
        const float xl = (float)(tx * 16 + r + half * 8) * kBinS;
        const float ox = fmaxf(fminf(xl + kBinS, bb.y) - fmaxf(xl, bb.x), 0.f);
        accx[r] += ox * cc.x * oy;
        accy[r] += ox * cc.y * oy;
      }
    }
#endif
  }

  // plain stores: each element owned by exactly one lane of one wave
  float* outx = partial + ((size_t)(chunk * 2 + 0) << 14);
  float* outy = partial + ((size_t)(chunk * 2 + 1) << 14);
#pragma unroll
  for (int r = 0; r < 8; ++r) {
    const int row = tx * 16 + r + half * 8;   // x-bin
    const int col = ty * 16 + lm;             // y-bin
    outx[row * kNumBins + col] = accx[r];
    outy[row * kNumBins + col] = accy[r];
  }
}

// ---------------------------------------------------------------------------
// Kernel 3: reduce chunks in fixed order + utilization clip
// ---------------------------------------------------------------------------
__global__ __launch_bounds__(256) void util_kernel(
    const float* __restrict__ partial,
    float*       __restrict__ util) {
  const int t = blockIdx.x * blockDim.x + threadIdx.x;
  if (t >= kMapElems) return;
  float sx = 0.f, sy = 0.f;
  for (int c = 0; c < kChunks; ++c) {
    sx += partial[((size_t)(c * 2 + 0) << 14) + t];
    sy += partial[((size_t)(c * 2 + 1) << 14) + t];
  }
  const float ux = sx * kInvHCap;
  const float uy = sy * kInvVCap;
  util[t] = fminf(fmaxf(fmaxf(ux, uy), 0.5f), 2.0f);
}

// ---------------------------------------------------------------------------
// Kernel 4: per-movable-node area.  util map (64 KB) staged into LDS with
// CDNA5 async global->LDS DMA (ASYNCcnt path); each node touches <= 3x3 bins.
// ---------------------------------------------------------------------------
__global__ __launch_bounds__(256) void instance_kernel(
    const float* __restrict__ pos,
    const float* __restrict__ node_size_x,
    const float* __restrict__ node_size_y,
    const float* __restrict__ util,
    float*       __restrict__ out) {
  __shared__ float s_util[kMapElems];   // 64 KB (WGP has 320 KB)

#if defined(__gfx1250__)
  // async copy: 4096 x 16B; 256 threads -> 16 per thread, tracked on ASYNCcnt
  for (int i = threadIdx.x; i < kMapElems / 4; i += 256) {
    const float4* gsrc = (const float4*)util + i;
    unsigned lds_addr = (unsigned)(unsigned long long)(s_util + 4 * i);
    asm volatile("global_load_async_to_lds_b128 %0, %1, off"
                 :
                 : "v"(lds_addr), "v"(gsrc)
                 : "memory");
  }
  asm volatile("s_wait_asynccnt 0x0" ::: "memory");
#else
  for (int i = threadIdx.x; i < kMapElems / 4; i += 256) {
    ((float4*)s_util)[i] = ((const float4*)util)[i];
  }
#endif
  __syncthreads();

  const int m = blockIdx.x * 256 + threadIdx.x;
  if (m >= kNumMovable) return;

  const float x0 = pos[m];
  const float y0 = pos[kNumNodes + m];
  const float x1 = x0 + node_size_x[m];
  const float y1 = y0 + node_size_y[m];

  int bx0 = (int)floorf(x0 / kBinS); bx0 = max(0, min(kNumBins - 1, bx0));
  int bx1 = (int)floorf(x1 / kBinS); bx1 = max(0, min(kNumBins - 1, bx1));
  int by0 = (int)floorf(y0 / kBinS); by0 = max(0, min(kNumBins - 1, by0));
  int by1 = (int)floorf(y1 / kBinS); by1 = max(0, min(kNumBins - 1, by1));

  float acc = 0.f;
  for (int ix = bx0; ix <= bx1; ++ix) {
    const float bl  = (float)ix * kBinS;
    const float oxv = fmaxf(fminf(bl + kBinS, x1) - fmaxf(bl, x0), 0.f);
    float rowacc = 0.f;
    for (int iy = by0; iy <= by1; ++iy) {
      const float b2  = (float)iy * kBinS;
      const float oyv = fmaxf(fminf(b2 + kBinS, y1) - fmaxf(b2, y0), 0.f);
      rowacc += oyv * s_util[ix * kNumBins + iy];
    }
    acc += oxv * rowacc;
  }
  out[m] = acc;
}

// ---------------------------------------------------------------------------
extern "C" void kernel_launch(void* const* d_in, const int* in_sizes, int n_in,
                              void* d_out, int out_size, void* d_ws, size_t ws_size,
                              hipStream_t stream) {
  const float* pos          = (const float*)d_in[0];
  const float* pin_pos      = (const float*)d_in[1];
  const float* node_size_x  = (const float*)d_in[2];
  const float* node_size_y  = (const float*)d_in[3];
  const float* net_weights  = (const float*)d_in[4];
  const int*   netpin_start = (const int*)d_in[5];
  const int*   flat_netpin  = (const int*)d_in[6];
  float*       out          = (float*)d_out;

  // workspace layout (floats): netf (padded) | partial maps | util (~11.7 MB)
  float* ws      = (float*)d_ws;
  float* netf    = ws;                                        // 8 * kPadNets
  float* partial = ws + (size_t)8 * kPadNets;                 // 64*2*16384
  float* util    = partial + (size_t)kChunks * 2 * kMapElems; // 16384

  hipLaunchKernelGGL(net_bbox_kernel, dim3((kPadNets + 255) / 256), dim3(256),
                     0, stream, pin_pos, netpin_start, flat_netpin, net_weights,
                     netf);
  hipLaunchKernelGGL(routemap_kernel, dim3(kChunks * 8), dim3(256), 0, stream,
                     netf, partial);
  hipLaunchKernelGGL(util_kernel, dim3(kMapElems / 256), dim3(256), 0, stream,
                     partial, util);
  hipLaunchKernelGGL(instance_kernel, dim3((kNumMovable + 255) / 256),
                     dim3(256), 0, stream, pos, node_size_x, node_size_y, util,
                     out);
}